// Net_9749575762300
// MI455X (gfx1250) — compile-verified
//
#include <hip/hip_runtime.h>
#include <math.h>

#define NNODES   50000
#define NEDGES   1600000
#define FIN      1433
#define HIDC     16
#define OUTC     7
#define ROWTILES 3125   // NNODES / 16 exactly
#define PAIRS    1563   // ceil(ROWTILES / 2)
#define FMAIN    1432   // 358 full K=4 steps; f=1432 handled in epilogue

typedef __attribute__((ext_vector_type(2))) float v2f;
typedef __attribute__((ext_vector_type(8))) float v8f;

// ---------------------------------------------------------------------------
// Layer-1 node transform on the WMMA pipe:
//   H1[n, k*16+o]  = sum_f x[n,f] * W1[k,f,o]      (cols 0..63)
//   XR1[n, o]      = sum_f x[n,f] * root1[f,o]     (cols 64..79)
// One wave owns a 32-row x 80-col tile (two 16-row WMMA tiles): the 5 B
// fragments per K-step are shared by both row tiles, so the loop runs
// 12 VMEM ops : 10 WMMAs. x streams from HBM exactly once; weights are
// re-read 1563x from L2 (458 KB slab). Main K-loop is guard-free; the
// single remainder column (f=1432) is a load-then-select epilogue.
// ---------------------------------------------------------------------------
__global__ __launch_bounds__(128) void gemm1_wmma(
    const float* __restrict__ x, const float* __restrict__ W1,
    const float* __restrict__ root1,
    float* __restrict__ h1, float* __restrict__ xr1)
{
    const int lane = threadIdx.x & 31;
    const int wave = threadIdx.x >> 5;
    const int pair = blockIdx.x * 4 + wave;
    if (pair >= PAIRS) return;                 // uniform per wave: EXEC stays all-1s

    const int rt0 = pair * 2;
    const int rt1 = (rt0 + 1 < ROWTILES) ? rt0 + 1 : rt0;  // clamp last odd tile
    const bool dup = (rt1 == rt0);

    const int m  = lane & 15;    // A: row within tile / B,D: col within tile
    const int kh = lane >> 4;    // 0 -> K pair {0,1}, 1 -> K pair {2,3}

    const float* xrow0 = x + (size_t)(rt0 * 16 + m) * FIN;
    const float* xrow1 = x + (size_t)(rt1 * 16 + m) * FIN;

    // B-matrix column-tile base pointers (stride 16 floats between f rows)
    const float* btab[5];
    btab[0] = W1 + 0 * FIN * HIDC + m;
    btab[1] = W1 + 1 * FIN * HIDC + m;
    btab[2] = W1 + 2 * FIN * HIDC + m;
    btab[3] = W1 + 3 * FIN * HIDC + m;
    btab[4] = root1 + m;

    v8f acc0[5] = {v8f{}, v8f{}, v8f{}, v8f{}, v8f{}};
    v8f acc1[5] = {v8f{}, v8f{}, v8f{}, v8f{}, v8f{}};

    // ---- main loop: f in [0, 1432), every load in-bounds, no guards ----
    #pragma unroll 2
    for (int f0 = 0; f0 < FMAIN; f0 += 4) {
        const int f = f0 + 2 * kh;             // max f+1 = 1431 < 1433
        v2f a0, a1;
        a0.x = xrow0[f];
        a0.y = xrow0[f + 1];
        a1.x = xrow1[f];
        a1.y = xrow1[f + 1];
        #pragma unroll
        for (int t = 0; t < 5; ++t) {
            v2f b;
            b.x = btab[t][(size_t)f * HIDC];
            b.y = btab[t][(size_t)f * HIDC + HIDC];
            acc0[t] = __builtin_amdgcn_wmma_f32_16x16x4_f32(
                false, a0, false, b, (short)0, acc0[t], false, false);
            acc1[t] = __builtin_amdgcn_wmma_f32_16x16x4_f32(
                false, a1, false, b, (short)0, acc1[t], false, false);
        }
    }

    // ---- epilogue: f = 1432 (K=1). Loads are in-bounds for ALL lanes;
    // zero the kh==1 half via select (v_cndmask, no EXEC branches).
    {
        const float ax0 = xrow0[FMAIN];
        const float ax1 = xrow1[FMAIN];
        v2f a0, a1;
        a0.x = (kh == 0) ? ax0 : 0.0f;  a0.y = 0.0f;
        a1.x = (kh == 0) ? ax1 : 0.0f;  a1.y = 0.0f;
        #pragma unroll
        for (int t = 0; t < 5; ++t) {
            const float bx = btab[t][(size_t)FMAIN * HIDC];
            v2f b;
            b.x = (kh == 0) ? bx : 0.0f;
            b.y = 0.0f;
            acc0[t] = __builtin_amdgcn_wmma_f32_16x16x4_f32(
                false, a0, false, b, (short)0, acc0[t], false, false);
            acc1[t] = __builtin_amdgcn_wmma_f32_16x16x4_f32(
                false, a1, false, b, (short)0, acc1[t], false, false);
        }
    }

    // D layout: VGPR v holds row (v + 8*kh), col m
    {
        const int nodeBase = rt0 * 16 + 8 * kh;
        #pragma unroll
        for (int v = 0; v < 8; ++v) {
            const size_t node = (size_t)(nodeBase + v);
            float* hrow = h1 + node * 64;
            hrow[ 0 + m] = acc0[0][v];
            hrow[16 + m] = acc0[1][v];
            hrow[32 + m] = acc0[2][v];
            hrow[48 + m] = acc0[3][v];
            xr1[node * 16 + m] = acc0[4][v];
        }
    }
    if (!dup) {                                // uniform per wave
        const int nodeBase = rt1 * 16 + 8 * kh;
        #pragma unroll
        for (int v = 0; v < 8; ++v) {
            const size_t node = (size_t)(nodeBase + v);
            float* hrow = h1 + node * 64;
            hrow[ 0 + m] = acc1[0][v];
            hrow[16 + m] = acc1[1][v];
            hrow[32 + m] = acc1[2][v];
            hrow[48 + m] = acc1[3][v];
            xr1[node * 16 + m] = acc1[4][v];
        }
    }
}

// ---------------------------------------------------------------------------
// Degree count (dst side)
// ---------------------------------------------------------------------------
__global__ void deg_kernel(const int* __restrict__ ei, float* __restrict__ deg)
{
    int e = blockIdx.x * blockDim.x + threadIdx.x;
    if (e >= NEDGES) return;
    atomicAdd(&deg[ei[NEDGES + e]], 1.0f);
}

__device__ __forceinline__ void edge_basis(const float* __restrict__ pseudo,
                                           int e, float b[4])
{
    float u = pseudo[2 * e + 0];
    float v = pseudo[2 * e + 1];
    b[0] = (1.0f - u) * (1.0f - v);
    b[1] = (1.0f - u) * v;
    b[2] = u * (1.0f - v);
    b[3] = u * v;
}

// Edge pass 1: msg[o] = sum_k basis[k] * H1[src, k*16+o]; scatter-add to agg1[dst]
__global__ void edge_msg1(const int* __restrict__ ei,
                          const float* __restrict__ pseudo,
                          const float* __restrict__ h1,
                          float* __restrict__ agg1)
{
    int e = blockIdx.x * blockDim.x + threadIdx.x;
    if (e >= NEDGES) return;
    int s = ei[e];
    int d = ei[NEDGES + e];
    float b[4];
    edge_basis(pseudo, e, b);
    const float* hs = h1 + (size_t)s * 64;   // contiguous 256B row per edge
    float* ad = agg1 + (size_t)d * 16;
    #pragma unroll
    for (int o = 0; o < 16; ++o) {
        float msg = b[0] * hs[o] + b[1] * hs[16 + o] +
                    b[2] * hs[32 + o] + b[3] * hs[48 + o];
        atomicAdd(&ad[o], msg);
    }
}

// Finish layer 1 (mean-agg + root + ELU) and do the tiny layer-2 node transform
__global__ void node_mid(const float* __restrict__ agg1,
                         const float* __restrict__ xr1,
                         const float* __restrict__ deg,
                         const float* __restrict__ W2,
                         const float* __restrict__ root2,
                         float* __restrict__ h2,
                         float* __restrict__ xr2)
{
    int n = blockIdx.x * blockDim.x + threadIdx.x;
    if (n >= NNODES) return;
    float dg  = deg[n];
    float inv = (dg > 0.0f) ? (1.0f / dg) : 0.0f;

    float h[16];
    #pragma unroll
    for (int o = 0; o < 16; ++o) {
        float a = agg1[(size_t)n * 16 + o] * inv + xr1[(size_t)n * 16 + o];
        h[o] = (a > 0.0f) ? a : expm1f(a);          // ELU(alpha=1)
    }

    float* h2r = h2 + (size_t)n * 28;
    #pragma unroll
    for (int k = 0; k < 4; ++k) {
        #pragma unroll
        for (int o2 = 0; o2 < 7; ++o2) {
            float acc = 0.0f;
            #pragma unroll
            for (int o = 0; o < 16; ++o)
                acc += h[o] * W2[k * (16 * 7) + o * 7 + o2];
            h2r[k * 7 + o2] = acc;
        }
    }
    #pragma unroll
    for (int o2 = 0; o2 < 7; ++o2) {
        float acc = 0.0f;
        #pragma unroll
        for (int o = 0; o < 16; ++o)
            acc += h[o] * root2[o * 7 + o2];
        xr2[(size_t)n * 7 + o2] = acc;
    }
}

// Edge pass 2: 7-wide messages from h2[src], scatter-add to agg2[dst]
__global__ void edge_msg2(const int* __restrict__ ei,
                          const float* __restrict__ pseudo,
                          const float* __restrict__ h2,
                          float* __restrict__ agg2)
{
    int e = blockIdx.x * blockDim.x + threadIdx.x;
    if (e >= NEDGES) return;
    int s = ei[e];
    int d = ei[NEDGES + e];
    float b[4];
    edge_basis(pseudo, e, b);
    const float* hs = h2 + (size_t)s * 28;
    float* ad = agg2 + (size_t)d * 7;
    #pragma unroll
    for (int o = 0; o < 7; ++o) {
        float msg = b[0] * hs[o] + b[1] * hs[7 + o] +
                    b[2] * hs[14 + o] + b[3] * hs[21 + o];
        atomicAdd(&ad[o], msg);
    }
}

// Final: mean-agg + root, then log_softmax over 7 classes
__global__ void node_out(const float* __restrict__ agg2,
                         const float* __restrict__ xr2,
                         const float* __restrict__ deg,
                         float* __restrict__ out)
{
    int n = blockIdx.x * blockDim.x + threadIdx.x;
    if (n >= NNODES) return;
    float dg  = deg[n];
    float inv = (dg > 0.0f) ? (1.0f / dg) : 0.0f;

    float z[7];
    float zmax = -1e30f;
    #pragma unroll
    for (int j = 0; j < 7; ++j) {
        z[j] = agg2[(size_t)n * 7 + j] * inv + xr2[(size_t)n * 7 + j];
        zmax = fmaxf(zmax, z[j]);
    }
    float ssum = 0.0f;
    #pragma unroll
    for (int j = 0; j < 7; ++j) ssum += expf(z[j] - zmax);
    float lse = zmax + logf(ssum);
    #pragma unroll
    for (int j = 0; j < 7; ++j) out[(size_t)n * 7 + j] = z[j] - lse;
}

// ---------------------------------------------------------------------------
extern "C" void kernel_launch(void* const* d_in, const int* in_sizes, int n_in,
                              void* d_out, int out_size, void* d_ws, size_t ws_size,
                              hipStream_t stream)
{
    const float* x     = (const float*)d_in[0];
    const int*   ei    = (const int*)  d_in[1];
    const float* ps    = (const float*)d_in[2];
    const float* W1    = (const float*)d_in[3];
    const float* root1 = (const float*)d_in[4];
    const float* W2    = (const float*)d_in[5];
    const float* root2 = (const float*)d_in[6];
    float* out = (float*)d_out;

    // Workspace layout (floats). deg/agg1/agg2 contiguous at front -> one memset.
    float* ws   = (float*)d_ws;
    float* deg  = ws;                         // N
    float* agg1 = ws + (size_t)NNODES;        // N*16
    float* agg2 = ws + (size_t)NNODES * 17;   // N*7
    float* h1   = ws + (size_t)NNODES * 24;   // N*64
    float* xr1  = ws + (size_t)NNODES * 88;   // N*16
    float* h2   = ws + (size_t)NNODES * 104;  // N*28
    float* xr2  = ws + (size_t)NNODES * 132;  // N*7   (total N*139 floats ~ 28 MB)

    hipMemsetAsync(ws, 0, (size_t)NNODES * 24 * sizeof(float), stream);

    const int EB = 256;
    const int eblocks = (NEDGES + EB - 1) / EB;
    const int nblocks = (NNODES + EB - 1) / EB;

    deg_kernel<<<eblocks, EB, 0, stream>>>(ei, deg);
    gemm1_wmma<<<(PAIRS + 3) / 4, 128, 0, stream>>>(x, W1, root1, h1, xr1);
    edge_msg1<<<eblocks, EB, 0, stream>>>(ei, ps, h1, agg1);
    node_mid<<<nblocks, EB, 0, stream>>>(agg1, xr1, deg, W2, root2, h2, xr2);
    edge_msg2<<<eblocks, EB, 0, stream>>>(ei, ps, h2, agg2);
    node_out<<<nblocks, EB, 0, stream>>>(agg2, xr2, deg, out);
}